// RGCNEncoder_9302899163872
// MI455X (gfx1250) — compile-verified
//
#include <hip/hip_runtime.h>

typedef float v2f __attribute__((ext_vector_type(2)));
typedef float v8f __attribute__((ext_vector_type(8)));

#define HID 128
#define N_AUTHOR 100000
#define N_PAPER  200000
#define N_EDGE   1000000

// ---------------------------------------------------------------------------
// Count in-degree (float, matches reference segment_sum of ones)
// ---------------------------------------------------------------------------
__global__ __launch_bounds__(256) void count_kernel(const int* __restrict__ dst,
                                                    float* __restrict__ cnt,
                                                    int n_edges) {
  int e = blockIdx.x * blockDim.x + threadIdx.x;
  if (e < n_edges) atomicAdd(&cnt[dst[e]], 1.0f);
}

// cnt -> 1 / max(cnt, 1)
__global__ __launch_bounds__(256) void inv_kernel(float* __restrict__ buf, int n) {
  int i = blockIdx.x * blockDim.x + threadIdx.x;
  if (i < n) buf[i] = 1.0f / fmaxf(buf[i], 1.0f);
}

// ---------------------------------------------------------------------------
// Scatter-add: one wave32 per edge; 512B coalesced gather, 4 f32 atomics/lane
// ---------------------------------------------------------------------------
__global__ __launch_bounds__(256) void scatter_kernel(const float* __restrict__ hsrc,
                                                      const int* __restrict__ src,
                                                      const int* __restrict__ dst,
                                                      float* __restrict__ sums,
                                                      int n_edges) {
  int e    = (int)((blockIdx.x * blockDim.x + threadIdx.x) >> 5);
  int lane = threadIdx.x & 31;
  if (e >= n_edges) return;
  int s = src[e];
  int d = dst[e];
  const float4 v = *(const float4*)(hsrc + (size_t)s * HID + lane * 4);
  float* p = sums + (size_t)d * HID + lane * 4;
  atomicAdd(p + 0, v.x);
  atomicAdd(p + 1, v.y);
  atomicAdd(p + 2, v.z);
  atomicAdd(p + 3, v.w);
}

// ---------------------------------------------------------------------------
// Projection GEMM: out[M,128] = X[M,256] @ W[128,256]^T   (f32 WMMA 16x16x4)
// Block: 64 rows x 128 cols; 8 waves, each wave owns one 16-col N-tile across
// 4 M-tiles (B fragment reused in registers 4x per K-step).
// M (=200000) must be a multiple of 64.
// ---------------------------------------------------------------------------
__global__ __launch_bounds__(256) void gemm_proj_kernel(const float* __restrict__ X,
                                                        const float* __restrict__ W,
                                                        float* __restrict__ out) {
  __shared__ float Atile[64][260];  // stride 260 % 64 == 4 -> conflict-free
  const int row0 = blockIdx.x * 64;
  const int tid  = threadIdx.x;

  // stage A tile: 64 rows x 256 cols f32
  for (int i = tid; i < 64 * 64; i += 256) {
    int r  = i >> 6;
    int c4 = (i & 63) * 4;
    *(float4*)&Atile[r][c4] = *(const float4*)(X + (size_t)(row0 + r) * 256 + c4);
  }
  __syncthreads();

  const int lane = tid & 31;
  const int n0   = (tid >> 5) * 16;       // wave's N tile
  const int mrow = lane & 15;             // A row within tile / B col
  const int koff = (lane >> 4) * 2;       // k-pair select per ISA layout
  const int n    = n0 + mrow;

  v8f acc[4] = {};
  #pragma unroll 8
  for (int kk = 0; kk < 256; kk += 4) {
    v2f b;
    b.x = W[(size_t)n * 256 + kk + koff];
    b.y = W[(size_t)n * 256 + kk + koff + 1];
    #pragma unroll
    for (int t = 0; t < 4; t++) {
      v2f a;
      a.x = Atile[t * 16 + mrow][kk + koff];
      a.y = Atile[t * 16 + mrow][kk + koff + 1];
      acc[t] = __builtin_amdgcn_wmma_f32_16x16x4_f32(false, a, false, b, (short)0,
                                                     acc[t], false, false);
    }
  }

  #pragma unroll
  for (int t = 0; t < 4; t++) {
    #pragma unroll
    for (int i = 0; i < 8; i++) {
      int m = t * 16 + i + ((lane >> 4) << 3);
      out[(size_t)(row0 + m) * HID + n0 + (lane & 15)] = acc[t][i];
    }
  }
}

// ---------------------------------------------------------------------------
// Fused SAGE combine:
//   expr = (sums*inv) @ Wl^T + bl + hdst @ Wr^T
//   v = (expr [+ out]) * scale ; [relu] ; out = v
// flags: bit0 = add previous out, bit1 = relu
// Block: 64 rows x 128 cols; 8 waves, one N-tile each across 4 M-tiles.
// Row-guarded (M need not divide 64; M must be a multiple of 16 -- it is).
// ---------------------------------------------------------------------------
__global__ __launch_bounds__(256) void sage_combine_kernel(
    const float* __restrict__ sums, const float* __restrict__ inv,
    const float* __restrict__ hdst, const float* __restrict__ Wl,
    const float* __restrict__ bl, const float* __restrict__ Wr,
    float* __restrict__ out, int M, int flags, float scale) {
  __shared__ float Agg[64][132];  // stride 132 % 64 == 4 -> conflict-free
  __shared__ float Hd[64][132];
  const int row0 = blockIdx.x * 64;
  const int tid  = threadIdx.x;

  // stage agg (mean) and root tiles: 64 x 128 each, row-guarded
  for (int i = tid; i < 64 * 32; i += 256) {
    int r  = i >> 5;
    int c4 = (i & 31) * 4;
    if (row0 + r < M) {
      float  iv = inv[row0 + r];
      float4 sv = *(const float4*)(sums + (size_t)(row0 + r) * HID + c4);
      sv.x *= iv; sv.y *= iv; sv.z *= iv; sv.w *= iv;
      *(float4*)&Agg[r][c4] = sv;
      *(float4*)&Hd[r][c4]  = *(const float4*)(hdst + (size_t)(row0 + r) * HID + c4);
    }
  }
  __syncthreads();

  const int lane = tid & 31;
  const int n0   = (tid >> 5) * 16;
  const int mrow = lane & 15;
  const int koff = (lane >> 4) * 2;
  const int n    = n0 + mrow;

  v8f acc[4] = {};
  #pragma unroll
  for (int kk = 0; kk < HID; kk += 4) {
    v2f b1, b2;
    b1.x = Wl[(size_t)n * HID + kk + koff];
    b1.y = Wl[(size_t)n * HID + kk + koff + 1];
    b2.x = Wr[(size_t)n * HID + kk + koff];
    b2.y = Wr[(size_t)n * HID + kk + koff + 1];
    #pragma unroll
    for (int t = 0; t < 4; t++) {
      v2f a1, a2;
      a1.x = Agg[t * 16 + mrow][kk + koff];
      a1.y = Agg[t * 16 + mrow][kk + koff + 1];
      acc[t] = __builtin_amdgcn_wmma_f32_16x16x4_f32(false, a1, false, b1, (short)0,
                                                     acc[t], false, false);
      a2.x = Hd[t * 16 + mrow][kk + koff];
      a2.y = Hd[t * 16 + mrow][kk + koff + 1];
      acc[t] = __builtin_amdgcn_wmma_f32_16x16x4_f32(false, a2, false, b2, (short)0,
                                                     acc[t], false, false);
    }
  }

  const float bias = bl[n0 + (lane & 15)];
  #pragma unroll
  for (int t = 0; t < 4; t++) {
    #pragma unroll
    for (int i = 0; i < 8; i++) {
      int m = t * 16 + i + ((lane >> 4) << 3);
      if (row0 + m < M) {
        size_t idx = (size_t)(row0 + m) * HID + n0 + (lane & 15);
        float  v   = acc[t][i] + bias;
        if (flags & 1) v += out[idx];
        v *= scale;
        if (flags & 2) v = fmaxf(v, 0.0f);
        out[idx] = v;
      }
    }
  }
}

// ---------------------------------------------------------------------------
extern "C" void kernel_launch(void* const* d_in, const int* in_sizes, int n_in,
                              void* d_out, int out_size, void* d_ws, size_t ws_size,
                              hipStream_t stream) {
  (void)in_sizes; (void)n_in; (void)out_size; (void)ws_size;

  const float* x_author = (const float*)d_in[0];
  const float* x_paper  = (const float*)d_in[1];
  const int*   w_src    = (const int*)d_in[2];
  const int*   w_dst    = (const int*)d_in[3];
  const int*   b_src    = (const int*)d_in[4];
  const int*   b_dst    = (const int*)d_in[5];
  const int*   c_src    = (const int*)d_in[6];
  const int*   c_dst    = (const int*)d_in[7];
  const float* linW     = (const float*)d_in[8];
  const float* Wl       = (const float*)d_in[9];
  const float* bl       = (const float*)d_in[10];
  const float* Wr       = (const float*)d_in[11];

  const size_t NP = N_PAPER, NA = N_AUTHOR;

  float* ws     = (float*)d_ws;
  float* h_p0   = ws;                    // NP*128
  float* h_p1   = h_p0 + NP * HID;       // NP*128
  float* h_a1   = h_p1 + NP * HID;       // NA*128
  float* sums_p = h_a1 + NA * HID;       // NP*128
  float* sums_a = sums_p + NP * HID;     // NA*128
  float* inv_w  = sums_a + NA * HID;     // NP  (writes -> paper degrees)
  float* inv_b  = inv_w + NP;            // NA  (written_by -> author degrees)
  float* inv_c  = inv_b + NA;            // NP  (cites -> paper degrees)

  const int gridP = (int)((NP + 63) / 64);
  const int gridA = (int)((NA + 63) / 64);

  // --- degree reciprocals (index-only; computed once, reused both layers) ---
  size_t inv_total = 2 * NP + NA;
  hipMemsetAsync(inv_w, 0, inv_total * sizeof(float), stream);
  count_kernel<<<(N_EDGE + 255) / 256, 256, 0, stream>>>(w_dst, inv_w, N_EDGE);
  count_kernel<<<(N_EDGE + 255) / 256, 256, 0, stream>>>(b_dst, inv_b, N_EDGE);
  count_kernel<<<(N_EDGE + 255) / 256, 256, 0, stream>>>(c_dst, inv_c, N_EDGE);
  inv_kernel<<<((int)inv_total + 255) / 256, 256, 0, stream>>>(inv_w, (int)inv_total);

  // --- input projection: h_p0 = x_paper @ linW^T  (200000 % 64 == 0) ---
  gemm_proj_kernel<<<(int)(NP / 64), 256, 0, stream>>>(x_paper, linW, h_p0);

  const float* ha_in = x_author;
  const float* hp_in = h_p0;

  for (int l = 0; l < 2; l++) {
    float* ha_out = (l == 0) ? h_a1 : (float*)d_out;
    float* hp_out = (l == 0) ? h_p1 : ((float*)d_out + NA * HID);

    const float* wl0 = Wl + (size_t)(l * 3 + 0) * HID * HID;
    const float* wl1 = Wl + (size_t)(l * 3 + 1) * HID * HID;
    const float* wl2 = Wl + (size_t)(l * 3 + 2) * HID * HID;
    const float* wr0 = Wr + (size_t)(l * 3 + 0) * HID * HID;
    const float* wr1 = Wr + (size_t)(l * 3 + 1) * HID * HID;
    const float* wr2 = Wr + (size_t)(l * 3 + 2) * HID * HID;
    const float* bl0 = bl + (size_t)(l * 3 + 0) * HID;
    const float* bl1 = bl + (size_t)(l * 3 + 1) * HID;
    const float* bl2 = bl + (size_t)(l * 3 + 2) * HID;

    // etype 0: author -writes-> paper  (partial result, no activation yet)
    hipMemsetAsync(sums_p, 0, NP * HID * sizeof(float), stream);
    scatter_kernel<<<N_EDGE / 8, 256, 0, stream>>>(ha_in, w_src, w_dst, sums_p, N_EDGE);
    sage_combine_kernel<<<gridP, 256, 0, stream>>>(
        sums_p, inv_w, hp_in, wl0, bl0, wr0, hp_out, (int)NP, /*flags=*/0, 1.0f);

    // etype 1: paper -written_by-> author  (h_a_new = relu(expr))
    hipMemsetAsync(sums_a, 0, NA * HID * sizeof(float), stream);
    scatter_kernel<<<N_EDGE / 8, 256, 0, stream>>>(hp_in, b_src, b_dst, sums_a, N_EDGE);
    sage_combine_kernel<<<gridA, 256, 0, stream>>>(
        sums_a, inv_b, ha_in, wl1, bl1, wr1, ha_out, (int)NA, /*flags=*/2, 1.0f);

    // etype 2: paper -cites-> paper  (h_p_new = relu(0.5*(prev + expr)))
    hipMemsetAsync(sums_p, 0, NP * HID * sizeof(float), stream);
    scatter_kernel<<<N_EDGE / 8, 256, 0, stream>>>(hp_in, c_src, c_dst, sums_p, N_EDGE);
    sage_combine_kernel<<<gridP, 256, 0, stream>>>(
        sums_p, inv_c, hp_in, wl2, bl2, wr2, hp_out, (int)NP, /*flags=*/3, 0.5f);

    ha_in = ha_out;
    hp_in = hp_out;
  }
}